// GCN_31164282699921
// MI455X (gfx1250) — compile-verified
//
#include <hip/hip_runtime.h>

typedef float v2f __attribute__((ext_vector_type(2)));
typedef float v8f __attribute__((ext_vector_type(8)));

#define DCH 128
#define LN_EPS 1e-5f

// ---------------- degree / dinv precompute ----------------
__global__ void k_init_deg(float* __restrict__ deg, int n) {
    int i = blockIdx.x * 256 + threadIdx.x;
    if (i < n) deg[i] = 1.0f;   // self-loop contributes 1 to every node
}

__global__ void k_deg_accum(const int* __restrict__ dst, float* __restrict__ deg, int e) {
    int i = blockIdx.x * 256 + threadIdx.x;
    if (i < e) atomicAdd(&deg[dst[i]], 1.0f);
}

__global__ void k_dinv(float* __restrict__ deg, int n) {
    int i = blockIdx.x * 256 + threadIdx.x;
    if (i < n) deg[i] = rsqrtf(deg[i]);   // deg >= 1 always (self loop)
}

// ---------------- fp32 WMMA GEMM: Ys[N,128] = (H[N,128] @ W[128,128]) * dinv[row] ----
// One block per 16-row M tile; 8 waves, wave w owns 16-col N tile w.
// A tile staged in LDS with padded stride (132 floats) for conflict-free reads.
// dinv[src] is folded into the output so the edge scatter is a pure add.
__global__ __launch_bounds__(256) void k_gemm(const float* __restrict__ H,
                                              const float* __restrict__ W,
                                              const float* __restrict__ dinv,
                                              float* __restrict__ Ys, int n) {
    __shared__ float ldsA[16 * 132];
    const int tid  = threadIdx.x;
    const int wave = tid >> 5;
    const int lane = tid & 31;
    const int row0 = blockIdx.x * 16;

    // stage A tile (16 x 128 floats) via float4, 2 per thread
    for (int i = tid; i < 512; i += 256) {
        int r  = i >> 5;       // 0..15
        int c4 = i & 31;       // 0..31 (float4 column)
        int gr = row0 + r; if (gr >= n) gr = n - 1;
        float4 v = *(const float4*)(H + (size_t)gr * DCH + c4 * 4);
        *(float4*)(&ldsA[r * 132 + c4 * 4]) = v;
    }
    __syncthreads();

    const int col0 = wave * 16;
    const int m    = lane & 15;          // A row / B,C column index within tile
    const int kb   = (lane >> 4) * 2;    // K sub-offset per ISA f32 16x4 layout
    const int ncol = col0 + m;           // global output column for this lane

    v8f c = {0.f, 0.f, 0.f, 0.f, 0.f, 0.f, 0.f, 0.f};
#pragma unroll 8
    for (int kk = 0; kk < DCH; kk += 4) {
        // A 16x4: v0=K+kb, v1=K+kb+1  (lanes 0-15: kb=0, lanes 16-31: kb=2)
        v2f a = *(const v2f*)(&ldsA[m * 132 + kk + kb]);
        // B 4x16: same K split across lane halves, row-striped across lanes
        v2f b;
        b.x = W[(size_t)(kk + kb) * DCH + ncol];
        b.y = W[(size_t)(kk + kb + 1) * DCH + ncol];
        c = __builtin_amdgcn_wmma_f32_16x16x4_f32(false, a, false, b,
                                                  (short)0, c, false, false);
    }

    // C/D 16x16 f32 layout: lanes 0-15 -> M=r, lanes 16-31 -> M=r+8
    const int mbase = (lane >> 4) * 8;
    if (row0 + 16 <= n) {
        // uniform fast path: no per-row exec juggling
#pragma unroll
        for (int r = 0; r < 8; ++r) {
            int grow = row0 + mbase + r;
            Ys[(size_t)grow * DCH + ncol] = c[r] * dinv[grow];
        }
    } else {
#pragma unroll
        for (int r = 0; r < 8; ++r) {
            int grow = row0 + mbase + r;
            if (grow < n) Ys[(size_t)grow * DCH + ncol] = c[r] * dinv[grow];
        }
    }
}

// ---------------- edge scatter: agg[dst] += Ys[src] ----------------
// One wave per edge; lane moves a float4 and issues 4 f32 atomic adds.
// (dinv factors are folded into Ys and the epilogue, self loops into epilogue.)
__global__ __launch_bounds__(256) void k_scatter(const float* __restrict__ Ys,
                                                 const int* __restrict__ src,
                                                 const int* __restrict__ dst,
                                                 float* __restrict__ agg,
                                                 int e_cnt) {
    int e    = blockIdx.x * 8 + (threadIdx.x >> 5);
    int lane = threadIdx.x & 31;
    if (e >= e_cnt) return;
    int s = src[e];
    int d = dst[e];
    float4 v = *(const float4*)(Ys + (size_t)s * DCH + lane * 4);
    float* o = agg + (size_t)d * DCH + lane * 4;
    atomicAdd(o + 0, v.x);
    atomicAdd(o + 1, v.y);
    atomicAdd(o + 2, v.z);
    atomicAdd(o + 3, v.w);
}

// -------- self-loop + dinv[dst] + bias + LayerNorm + PReLU: one wave per node --------
__global__ __launch_bounds__(256) void k_epilogue(const float* __restrict__ agg,
                                                  const float* __restrict__ Ys,
                                                  const float* __restrict__ dinv,
                                                  const float* __restrict__ bias,
                                                  const float* __restrict__ lnw,
                                                  const float* __restrict__ lnb,
                                                  const float* __restrict__ alphas,
                                                  int layer, float* __restrict__ out,
                                                  int n) {
    int node = blockIdx.x * 8 + (threadIdx.x >> 5);
    int lane = threadIdx.x & 31;
    if (node >= n) return;
    const float alpha = alphas[layer];
    const float dn    = dinv[node];
    const int c0 = lane * 4;
    float4 a4 = *(const float4*)(agg + (size_t)node * DCH + c0);
    float4 y4 = *(const float4*)(Ys  + (size_t)node * DCH + c0);  // self loop
    float v[4] = {a4.x + y4.x, a4.y + y4.y, a4.z + y4.z, a4.w + y4.w};
#pragma unroll
    for (int j = 0; j < 4; ++j) v[j] = v[j] * dn + bias[layer * DCH + c0 + j];

    float s = v[0] + v[1] + v[2] + v[3];
    float q = v[0]*v[0] + v[1]*v[1] + v[2]*v[2] + v[3]*v[3];
#pragma unroll
    for (int off = 16; off >= 1; off >>= 1) {
        s += __shfl_xor(s, off, 32);
        q += __shfl_xor(q, off, 32);
    }
    float mu  = s * (1.0f / DCH);
    float var = q * (1.0f / DCH) - mu * mu;
    float rin = rsqrtf(var + LN_EPS);

    float o[4];
#pragma unroll
    for (int j = 0; j < 4; ++j) {
        float t = (v[j] - mu) * rin;
        t = t * lnw[layer * DCH + c0 + j] + lnb[layer * DCH + c0 + j];
        o[j] = (t >= 0.0f) ? t : alpha * t;
    }
    *(float4*)(out + (size_t)node * DCH + c0) = make_float4(o[0], o[1], o[2], o[3]);
}

// ---------------- host-side orchestration ----------------
extern "C" void kernel_launch(void* const* d_in, const int* in_sizes, int n_in,
                              void* d_out, int out_size, void* d_ws, size_t ws_size,
                              hipStream_t stream) {
    const float* x      = (const float*)d_in[0];
    const int*   ei     = (const int*)  d_in[1];
    const float* Ws     = (const float*)d_in[2];
    const float* bs     = (const float*)d_in[3];
    const float* lnw    = (const float*)d_in[4];
    const float* lnb    = (const float*)d_in[5];
    const float* alphas = (const float*)d_in[6];

    const int N_ = in_sizes[0] / DCH;
    const int E_ = in_sizes[1] / 2;
    const int* srcA = ei;        // edge_index[0]
    const int* dstA = ei + E_;   // edge_index[1]

    size_t nd = (size_t)N_ * DCH;
    float* h    = (float*)d_ws;   // [N,128]
    float* ys   = h + nd;         // [N,128]  (H@W)*dinv[row]
    float* agg  = ys + nd;        // [N,128]
    float* dinv = agg + nd;       // [N]  (holds deg during precompute)

    // symmetric normalization factors (once)
    k_init_deg <<<(N_ + 255) / 256, 256, 0, stream>>>(dinv, N_);
    k_deg_accum<<<(E_ + 255) / 256, 256, 0, stream>>>(dstA, dinv, E_);
    k_dinv     <<<(N_ + 255) / 256, 256, 0, stream>>>(dinv, N_);

    const int mblocks = (N_ + 15) / 16;
    const int eblocks = (E_ + 7) / 8;
    const int nblocks = (N_ + 7) / 8;

    const float* hin = x;
    for (int l = 0; l < 3; ++l) {
        k_gemm<<<mblocks, 256, 0, stream>>>(hin, Ws + (size_t)l * DCH * DCH, dinv, ys, N_);
        hipMemsetAsync(agg, 0, nd * sizeof(float), stream);
        k_scatter<<<eblocks, 256, 0, stream>>>(ys, srcA, dstA, agg, E_);
        float* outp = (l == 2) ? (float*)d_out : h;
        k_epilogue<<<nblocks, 256, 0, stream>>>(agg, ys, dinv, bs, lnw, lnb, alphas, l, outp, N_);
        hin = h;
    }
}